// GeneGINet_8022998909689
// MI455X (gfx1250) — compile-verified
//
#include <hip/hip_runtime.h>

// Problem constants (from reference)
#define B_   4
#define G_   2000
#define D_   300
#define L_   2
#define E_   256000
#define N_   (B_ * G_)          // 8000 nodes
#define D2_  (2 * D_)           // 600
#define CH_  (D_ / 4)           // 75 float4 chunks per D-row
#define BN_EPS 1e-5f

typedef __attribute__((ext_vector_type(2))) float v2f;
typedef __attribute__((ext_vector_type(8))) float v8f;

// f32 WMMA: D(16x16,f32) = A(16x4,f32) * B(4x16,f32) + C
// A layout: lanes 0-15 -> rows M=0..15, {VGPR0,VGPR1} = K={0,1}; lanes 16-31 same rows, K={2,3}
// B layout: lanes 0-15 -> cols N=0..15, {VGPR0,VGPR1} = K rows {0,1}; lanes 16-31, K rows {2,3}
// C/D layout: VGPR r, lanes 0-15 -> M=r, lanes 16-31 -> M=r+8
__device__ __forceinline__ v8f wmma4(v2f a, v2f b, v8f c) {
  return __builtin_amdgcn_wmma_f32_16x16x4_f32(false, a, false, b, (short)0, c,
                                               false, false);
}

// Native non-returning f32 atomic add (global_atomic_add_f32)
__device__ __forceinline__ void atomAddF(float* p, float v) {
  __hip_atomic_fetch_add(p, v, __ATOMIC_RELAXED, __HIP_MEMORY_SCOPE_AGENT);
}

// ---------------------------------------------------------------------------
// Layer-invariant edge stats: deg[n] = #incoming edges, s[n] = sum of weights
// ---------------------------------------------------------------------------
__global__ void edge_stats_k(const long long* __restrict__ ei,
                             const float* __restrict__ ew,
                             float* __restrict__ deg, float* __restrict__ s) {
  int e = blockIdx.x * blockDim.x + threadIdx.x;
  if (e >= E_) return;
  int dst = (int)ei[(long long)E_ + e];
  atomAddF(&deg[dst], 1.0f);
  atomAddF(&s[dst], ew[e]);
}

// ---------------------------------------------------------------------------
// Scatter: agg[dst] += h[src]  (float4 gather, 4 native f32 atomics)
// thread = (edge, chunk); chunk fastest so a wave reads mostly one h-row
// ---------------------------------------------------------------------------
__global__ void scatter_k(const float* __restrict__ h,
                          const long long* __restrict__ ei,
                          float* __restrict__ agg) {
  long long tid = (long long)blockIdx.x * blockDim.x + threadIdx.x;
  if (tid >= (long long)E_ * CH_) return;
  int e = (int)(tid / CH_);
  int c = (int)(tid % CH_);
  int src = (int)ei[e];
  int dst = (int)ei[(long long)E_ + e];
  const float4 v = *(const float4*)(h + src * D_ + c * 4);
  float* ap = agg + dst * D_ + c * 4;
  atomAddF(ap + 0, v.x);
  atomAddF(ap + 1, v.y);
  atomAddF(ap + 2, v.z);
  atomAddF(ap + 3, v.w);
}

// agg[n,f] += deg[n]*b_edge[f] + s[n]*w_edge[f]   (collapsed edge embedding)
__global__ void finish_agg_k(float* __restrict__ agg,
                             const float* __restrict__ deg,
                             const float* __restrict__ s,
                             const float* __restrict__ be,
                             const float* __restrict__ we) {
  int tid = blockIdx.x * blockDim.x + threadIdx.x;
  if (tid >= N_ * CH_) return;
  int n = tid / CH_, c = tid % CH_;
  float dn = deg[n], sn = s[n];
  float4 a = *(float4*)(agg + n * D_ + c * 4);
  const float4 b4 = *(const float4*)(be + c * 4);
  const float4 w4 = *(const float4*)(we + c * 4);
  a.x += dn * b4.x + sn * w4.x;
  a.y += dn * b4.y + sn * w4.y;
  a.z += dn * b4.z + sn * w4.z;
  a.w += dn * b4.w + sn * w4.w;
  *(float4*)(agg + n * D_ + c * 4) = a;
}

// ---------------------------------------------------------------------------
// WMMA GEMM: C = act(A[MxK] * B[KxNn] + bias), one 16x64 strip per wave.
// K, Nn are compile-time so all B addressing strength-reduces to immediate
// offsets / constant pointer increments (b.y load = b.x address + Nn*4 imm).
// N-edge handling: OOB columns are *clamped* for the B loads (column n of D
// depends only on column n of B, same lane), so the inner loop is branch-free;
// garbage columns are dropped by the store-side guard. EXEC stays all-1s.
// ---------------------------------------------------------------------------
template <bool RELU, int K, int Nn>
__global__ void gemm_wmma_k(const float* __restrict__ A,
                            const float* __restrict__ Bm,
                            const float* __restrict__ bias,
                            float* __restrict__ C) {
  constexpr int tilesN64 = (Nn + 63) / 64;
  constexpr int totalWaves = (N_ / 16) * tilesN64;

  int wid = (int)((blockIdx.x * blockDim.x + threadIdx.x) >> 5);
  if (wid >= totalWaves) return;            // whole-wave uniform exit
  int lane = threadIdx.x & 31;
  int half = lane >> 4;                     // K sub-pair selector
  int lm = lane & 15;                       // row (A) / col (B,C) within tile
  int tm = wid / tilesN64;
  int tn = wid % tilesN64;
  int c0 = tn * 64 + lm;
  int c1 = c0 + 16, c2 = c0 + 32, c3 = c0 + 48;
  // clamped load columns (branch-free inner loop)
  int s0 = min(c0, Nn - 1);
  int s1 = min(c1, Nn - 1);
  int s2 = min(c2, Nn - 1);
  int s3 = min(c3, Nn - 1);

  const float* arow = A + (long long)(tm * 16 + lm) * K + half * 2;
  // per-column B base pointers for K rows {half*2, half*2+1}
  const float* bp0 = Bm + (long long)(half * 2) * Nn + s0;
  const float* bp1 = Bm + (long long)(half * 2) * Nn + s1;
  const float* bp2 = Bm + (long long)(half * 2) * Nn + s2;
  const float* bp3 = Bm + (long long)(half * 2) * Nn + s3;

  v8f acc0 = {0, 0, 0, 0, 0, 0, 0, 0};
  v8f acc1 = acc0, acc2 = acc0, acc3 = acc0;

#pragma unroll 4
  for (int k = 0; k < K; k += 4) {
    v2f a = *(const v2f*)(arow + k);        // A[row][k+2*half .. +1]
    const long long koff = (long long)k * Nn;   // constant-folded per unroll
    v2f b;
    b.x = bp0[koff];  b.y = bp0[koff + Nn];
    acc0 = wmma4(a, b, acc0);
    b.x = bp1[koff];  b.y = bp1[koff + Nn];
    acc1 = wmma4(a, b, acc1);
    b.x = bp2[koff];  b.y = bp2[koff + Nn];
    acc2 = wmma4(a, b, acc2);
    b.x = bp3[koff];  b.y = bp3[koff + Nn];
    acc3 = wmma4(a, b, acc3);
  }

  int mbase = tm * 16 + half * 8;
#define STORE_TILE(ACC, CC)                                         \
  if ((CC) < Nn) {                                                  \
    float bb = bias[CC];                                            \
    _Pragma("unroll") for (int r = 0; r < 8; ++r) {                 \
      float v = ACC[r] + bb;                                        \
      if (RELU) v = fmaxf(v, 0.f);                                  \
      C[(long long)(mbase + r) * Nn + (CC)] = v;                    \
    }                                                               \
  }
  STORE_TILE(acc0, c0)
  STORE_TILE(acc1, c1)
  STORE_TILE(acc2, c2)
  STORE_TILE(acc3, c3)
#undef STORE_TILE
}

// ---------------------------------------------------------------------------
// BatchNorm: per-feature mean/rstd over N rows (one block per feature)
// ---------------------------------------------------------------------------
__global__ void bn_reduce_k(const float* __restrict__ h,
                            float* __restrict__ mean, float* __restrict__ rstd) {
  __shared__ float ssum[256];
  __shared__ float ssq[256];
  int f = blockIdx.x;
  int t = threadIdx.x;
  float sum = 0.f, sq = 0.f;
  for (int n = t; n < N_; n += 256) {
    float v = h[(long long)n * D_ + f];
    sum += v;
    sq += v * v;
  }
  ssum[t] = sum; ssq[t] = sq;
  __syncthreads();
  for (int w = 128; w > 0; w >>= 1) {
    if (t < w) { ssum[t] += ssum[t + w]; ssq[t] += ssq[t + w]; }
    __syncthreads();
  }
  if (t == 0) {
    float m = ssum[0] * (1.0f / N_);
    float var = ssq[0] * (1.0f / N_) - m * m;
    mean[f] = m;
    rstd[f] = rsqrtf(var + BN_EPS);
  }
}

__global__ void bn_apply_k(const float* __restrict__ h,
                           const float* __restrict__ mean,
                           const float* __restrict__ rstd,
                           const float* __restrict__ gamma,
                           const float* __restrict__ beta,
                           float* __restrict__ out, int relu) {
  int tid = blockIdx.x * blockDim.x + threadIdx.x;
  if (tid >= N_ * CH_) return;
  int n = tid / CH_, c = tid % CH_;
  float4 v = *(const float4*)(h + n * D_ + c * 4);
  const float4 me = *(const float4*)(mean + c * 4);
  const float4 rs = *(const float4*)(rstd + c * 4);
  const float4 g = *(const float4*)(gamma + c * 4);
  const float4 bt = *(const float4*)(beta + c * 4);
  float4 r;
  r.x = (v.x - me.x) * rs.x * g.x + bt.x;
  r.y = (v.y - me.y) * rs.y * g.y + bt.y;
  r.z = (v.z - me.z) * rs.z * g.z + bt.z;
  r.w = (v.w - me.w) * rs.w * g.w + bt.w;
  if (relu) {
    r.x = fmaxf(r.x, 0.f); r.y = fmaxf(r.y, 0.f);
    r.z = fmaxf(r.z, 0.f); r.w = fmaxf(r.w, 0.f);
  }
  *(float4*)(out + n * D_ + c * 4) = r;
}

// ---------------------------------------------------------------------------
extern "C" void kernel_launch(void* const* d_in, const int* in_sizes, int n_in,
                              void* d_out, int out_size, void* d_ws,
                              size_t ws_size, hipStream_t stream) {
  const float* x          = (const float*)d_in[0];
  const long long* ei     = (const long long*)d_in[1];   // int64 per reference
  const float* ew         = (const float*)d_in[2];
  const float* w_edge     = (const float*)d_in[3];       // [L,1,D]
  const float* b_edge     = (const float*)d_in[4];       // [L,D]
  const float* w1         = (const float*)d_in[5];       // [L,D,2D]
  const float* b1         = (const float*)d_in[6];       // [L,2D]
  const float* w2         = (const float*)d_in[7];       // [L,2D,D]
  const float* b2         = (const float*)d_in[8];       // [L,D]
  const float* gamma      = (const float*)d_in[9];       // [L,D]
  const float* beta       = (const float*)d_in[10];      // [L,D]
  float* out = (float*)d_out;

  float* ws   = (float*)d_ws;
  float* hbuf = ws;                               // N*D
  float* agg  = ws + (size_t)N_ * D_;             // N*D  (also reused as h2)
  float* tbuf = ws + (size_t)2 * N_ * D_;         // N*2D
  float* deg  = ws + (size_t)4 * N_ * D_;         // N
  float* sbuf = deg + N_;                         // N
  float* mean = sbuf + N_;                        // D
  float* rstd = mean + D_;                        // D

  // layer-invariant edge stats
  hipMemsetAsync(deg, 0, 2 * N_ * sizeof(float), stream);
  edge_stats_k<<<(E_ + 255) / 256, 256, 0, stream>>>(ei, ew, deg, sbuf);

  const int nthreads = N_ * CH_;
  const long long sthreads = (long long)E_ * CH_;

  // wave/block counts for the two GEMM shapes
  const int waves1 = (N_ / 16) * ((D2_ + 63) / 64);   // 5000
  const int blocks1 = (waves1 * 32 + 255) / 256;
  const int waves2 = (N_ / 16) * ((D_ + 63) / 64);    // 2500
  const int blocks2 = (waves2 * 32 + 255) / 256;

  for (int l = 0; l < L_; ++l) {
    const float* hin = (l == 0) ? x : hbuf;

    hipMemsetAsync(agg, 0, (size_t)N_ * D_ * sizeof(float), stream);
    scatter_k<<<(int)((sthreads + 255) / 256), 256, 0, stream>>>(hin, ei, agg);
    finish_agg_k<<<(nthreads + 255) / 256, 256, 0, stream>>>(
        agg, deg, sbuf, b_edge + l * D_, w_edge + l * D_);

    // t = relu(agg @ w1 + b1):  M=N_, K=D_, Nn=2D
    gemm_wmma_k<true, D_, D2_><<<blocks1, 256, 0, stream>>>(
        agg, w1 + (size_t)l * D_ * D2_, b1 + l * D2_, tbuf);

    // h2 = t @ w2 + b2 (into agg): M=N_, K=2D, Nn=D
    gemm_wmma_k<false, D2_, D_><<<blocks2, 256, 0, stream>>>(
        tbuf, w2 + (size_t)l * D2_ * D_, b2 + l * D_, agg);

    bn_reduce_k<<<D_, 256, 0, stream>>>(agg, mean, rstd);
    float* o = (l == L_ - 1) ? out : hbuf;
    bn_apply_k<<<(nthreads + 255) / 256, 256, 0, stream>>>(
        agg, mean, rstd, gamma + l * D_, beta + l * D_, o,
        (l < L_ - 1) ? 1 : 0);
  }
}